// _DecisionTreeTS_14628658610622
// MI455X (gfx1250) — compile-verified
//
#include <hip/hip_runtime.h>

// Decision-tree inference for MI455X (gfx1250).
// Memory-bound: ~300 MB total traffic @ 23.3 TB/s ≈ 13 us floor; WMMA is
// inapplicable (zero matmul content). CDNA5 paths: global_load_async_to_lds_b128
// (ASYNCcnt DMA, NT-hinted so the 256 MB X stream does not evict the ~3.4 MB
// L2-resident tree tables), s_wait_asynccnt, bank-conflict-free padded LDS rows,
// non-temporal b128 output stores.

#define TPB   256          // 8 wave32 per block
#define NFEAT 64
#define ROWP  65           // padded row stride (floats) -> conflict-free gathers
#define NOUT  10

typedef __attribute__((ext_vector_type(4))) float v4f;   // native vector: OK for
                                                         // __builtin_nontemporal_*

// One async b128 copy. INST_OFFSET is added to BOTH the LDS address
// (VDST + IOFF) and the global address (SADDR + VADDR + IOFF), so the shared
// k*4096 component lives in the immediate and only the row-padding residual
// (k*64 bytes) is accumulated into the VDST register.
#define ACOPY(OFS)                                                         \
    asm volatile("global_load_async_to_lds_b128 %0, %1, %2 offset:" #OFS   \
                 " th:TH_LOAD_NT"                                          \
                 :: "v"(ldsAddr), "v"(gOff), "s"(gbase) : "memory");       \
    ldsAddr += 64u

__device__ __forceinline__ int traverse_step(
    int node, const float* __restrict__ row,
    const int* __restrict__ feature, const int* __restrict__ childL,
    const int* __restrict__ childR, const float* __restrict__ thresh)
{
    int   f = feature[node];
    int   l = childL[node];
    int   r = childR[node];
    float t = thresh[node];
    bool stay = ((l == -1) && (r == -1)) || (f == -2);
    float x = row[stay ? 0 : f];
    int nxt = (x <= t) ? l : r;
    return stay ? node : nxt;
}

__device__ __forceinline__ void leaf_probs(
    int node, const float* __restrict__ value, float v[NOUT])
{
    const float* vp = value + (size_t)node * NOUT;   // 8B-aligned, L2-hot
    float sum = 0.f;
#pragma unroll
    for (int j = 0; j < NOUT; ++j) { v[j] = vp[j]; sum += v[j]; }
    if (sum > 0.f) {
        float inv = 1.0f / sum;
#pragma unroll
        for (int j = 0; j < NOUT; ++j) v[j] *= inv;
    } else {
#pragma unroll
        for (int j = 0; j < NOUT; ++j) v[j] = 1.0f / NOUT;
    }
}

__global__ __launch_bounds__(TPB) void dtree_predict_kernel(
    const float* __restrict__ X,
    const float* __restrict__ thresh,
    const float* __restrict__ value,
    const int*   __restrict__ childL,
    const int*   __restrict__ childR,
    const int*   __restrict__ feature,
    float*       __restrict__ out,
    int nSamples, int depth)
{
    __shared__ float tile[TPB * ROWP];     // 66,560 B: 256 padded feature rows

    const int tid = threadIdx.x;
    const long long blockBase = (long long)blockIdx.x * TPB;

    if (blockBase + TPB <= (long long)nSamples) {
        // ================= full-block fast path (no guards) =================
        // Stage the 256x64 f32 X tile into LDS via async DMA. Chunk k of
        // thread tid covers global floats g = tid*4 + k*1024 (never crosses a
        // row), landing at padded LDS byte ((g>>6)*65 + (g&63))*4
        //   = base + (tid>>4)*260 + (tid*4&63)*4 + k*4160.
        const float* gbase = X + blockBase * NFEAT;
        unsigned ldsAddr = (unsigned)(size_t)(&tile[0])
                         + ((unsigned)(tid >> 4) * ROWP + ((tid * 4u) & (NFEAT - 1))) * 4u;
        unsigned gOff = (unsigned)(tid * 16);
        ACOPY(0);     ACOPY(4096);  ACOPY(8192);  ACOPY(12288);
        ACOPY(16384); ACOPY(20480); ACOPY(24576); ACOPY(28672);
        ACOPY(32768); ACOPY(36864); ACOPY(40960); ACOPY(45056);
        ACOPY(49152); ACOPY(53248); ACOPY(57344); ACOPY(61440);
        asm volatile("s_wait_asynccnt 0" ::: "memory");
        __syncthreads();   // every wave waited its own ASYNCcnt -> tile ready

        // ---- lockstep traversal: LDS feature gathers, L2-hot node tables ----
        const float* row = &tile[tid * ROWP];
        int node = 0;
        if (depth == 15) {
#pragma unroll
            for (int d = 0; d < 15; ++d)
                node = traverse_step(node, row, feature, childL, childR, thresh);
        } else {
            for (int d = 0; d < depth; ++d)
                node = traverse_step(node, row, feature, childL, childR, thresh);
        }

        float v[NOUT];
        leaf_probs(node, value, v);

        // ---- stage 256x10 outputs, then coalesced NT b128 stores ----
        __syncthreads();                 // done reading tile -> safe to reuse
#pragma unroll
        for (int j = 0; j < NOUT; ++j) tile[tid * NOUT + j] = v[j];
        __syncthreads();

        const v4f* lsrc = (const v4f*)tile;                 // 640 b128 chunks
        v4f* gdst = (v4f*)(out + blockBase * NOUT);         // 16B aligned
        __builtin_nontemporal_store(lsrc[tid],       &gdst[tid]);
        __builtin_nontemporal_store(lsrc[tid + 256], &gdst[tid + 256]);
        if (tid < 128)
            __builtin_nontemporal_store(lsrc[tid + 512], &gdst[tid + 512]);
    } else {
        // ================= rare tail path (partial block) =================
        const long long s = blockBase + tid;
        if (s < (long long)nSamples) {
            const float* row = X + s * NFEAT;     // direct global gathers
            int node = 0;
            for (int d = 0; d < depth; ++d)
                node = traverse_step(node, row, feature, childL, childR, thresh);
            float v[NOUT];
            leaf_probs(node, value, v);
#pragma unroll
            for (int j = 0; j < NOUT; ++j) out[s * NOUT + j] = v[j];
        }
    }
}

extern "C" void kernel_launch(void* const* d_in, const int* in_sizes, int n_in,
                              void* d_out, int out_size, void* d_ws, size_t ws_size,
                              hipStream_t stream) {
    const float* X      = (const float*)d_in[0];
    const float* thresh = (const float*)d_in[1];
    const float* value  = (const float*)d_in[2];
    const int*   childL = (const int*)d_in[3];
    const int*   childR = (const int*)d_in[4];
    const int*   feat   = (const int*)d_in[5];
    float* out = (float*)d_out;

    const int nSamples = in_sizes[0] / NFEAT;        // 1,048,576
    const int n_nodes  = in_sizes[1];                // 65,535
    // complete heap: n_nodes = 2^(depth+1)-1  ->  depth = log2(n_nodes+1)-1
    const int depth = (31 - __builtin_clz((unsigned)(n_nodes + 1))) - 1;   // 15

    const int nBlocks = (nSamples + TPB - 1) / TPB;  // 4096
    dtree_predict_kernel<<<nBlocks, TPB, 0, stream>>>(
        X, thresh, value, childL, childR, feat, out, nSamples, depth);
}